// AdaFilter_1_53188874994327
// MI455X (gfx1250) — compile-verified
//
#include <hip/hip_runtime.h>

// ---------------- problem constants (fixed by setup_inputs) ----------------
#define EPSF 1e-5f
constexpr int NN   = 2;
constexpr int NP   = 32;
constexpr int NPAN = NN * NP;          // 64 (sample,panel) pairs
constexpr int H    = 185;
constexpr int W    = 388;
constexpr int HW   = H * W;
// encoder stage sizes
constexpr int E1H = 46, E1W = 49;      // (185-3)/4+1, (388-3)/8+1
constexpr int E2H = 6,  E2W = 6;
// tiling for the heavy convs
constexpr int TS  = 16;                // output tile (16x16 pixels)
constexpr int GT  = TS + 2;            // intermediate tile (+halo1) = 18
constexpr int XT  = TS + 4;            // input tile (+halo2) = 20
constexpr int TILES_X = (W + TS - 1) / TS;   // 25
constexpr int TILES_Y = (H + TS - 1) / TS;   // 12
constexpr int WB = 305;                // dynamic-param vector length per panel
constexpr int NPIX = GT * GT;          // 324 intermediate pixels per tile
constexpr int NCHK = (NPIX + 15) / 16; // 21 pixel-chunks of 16

typedef __attribute__((ext_vector_type(2))) float v2f;
typedef __attribute__((ext_vector_type(8))) float v8f;

#if defined(__AMDGCN__) && __has_builtin(__builtin_amdgcn_wmma_f32_16x16x4_f32)
#define USE_WMMA4 1
#else
#define USE_WMMA4 0
#endif

// gfx1250 async global->LDS staging (ASYNCcnt tracked); guarded fallback.
// Prototype (from hipcc diagnostic): (addrspace(1) int*, addrspace(3) int*,
// imm offset, imm cpol).
#if defined(__AMDGCN__) && \
    __has_builtin(__builtin_amdgcn_global_load_async_to_lds_b32) && \
    __has_builtin(__builtin_amdgcn_s_wait_asynccnt)
#define HAVE_ASYNC_LDS 1
__device__ __forceinline__ void stage_b32(const float* gsrc, float* ldst) {
    __builtin_amdgcn_global_load_async_to_lds_b32(
        (__attribute__((address_space(1))) int*)gsrc,
        (__attribute__((address_space(3))) int*)ldst, 0, 0);
}
__device__ __forceinline__ void stage_wait() {
    __builtin_amdgcn_s_wait_asynccnt(0);
}
#else
#define HAVE_ASYNC_LDS 0
__device__ __forceinline__ void stage_b32(const float* gsrc, float* ldst) {
    *ldst = *gsrc;
}
__device__ __forceinline__ void stage_wait() {}
#endif

__device__ __forceinline__ int refl(int i, int n) {
    i = (i < 0) ? -i : i;
    return (i >= n) ? (2 * n - 2 - i) : i;
}
__device__ __forceinline__ int clampi(int v, int lo, int hi) {
    return v < lo ? lo : (v > hi ? hi : v);
}

// ============================================================================
// Kernel 1: panel encoder (3 grouped convs + GroupNorm + ReLU) + linear head.
// One workgroup per (sample,panel). Produces wb[panel][305] dynamic params.
// ============================================================================
__global__ __launch_bounds__(256) void encoder_kernel(
    const float* __restrict__ x,
    const float* __restrict__ w_e1, const float* __restrict__ b_e1,
    const float* __restrict__ g_e1, const float* __restrict__ be_e1,
    const float* __restrict__ w_e2, const float* __restrict__ b_e2,
    const float* __restrict__ g_e2, const float* __restrict__ be_e2,
    const float* __restrict__ w_e3, const float* __restrict__ b_e3,
    const float* __restrict__ g_e3, const float* __restrict__ be_e3,
    const float* __restrict__ W_lin, const float* __restrict__ b_lin,
    float* __restrict__ wb_ws)
{
    __shared__ float s1[2 * E1H * E1W];   // 4508
    __shared__ float s2[4 * E2H * E2W];   // 144
    __shared__ float s3[8];
    __shared__ float reds[256], redq[256];
    __shared__ float s_mean, s_rstd;

    const int bp  = blockIdx.x;           // 0..63   (n*32 + p)
    const int p   = bp % NP;
    const int tid = threadIdx.x;
    const float* xp = x + (size_t)bp * HW;

    // ---- stage e1: 1->2 ch, 3x3, stride (4,8), VALID ----
    float ls = 0.f, lq = 0.f;
    for (int idx = tid; idx < 2 * E1H * E1W; idx += 256) {
        int oc = idx / (E1H * E1W);
        int r  = idx % (E1H * E1W);
        int oy = r / E1W, ox = r % E1W;
        const float* wk = w_e1 + (2 * p + oc) * 9;
        float acc = b_e1[2 * p + oc];
        int iy = oy * 4, ix = ox * 8;
        #pragma unroll
        for (int ky = 0; ky < 3; ++ky)
            #pragma unroll
            for (int kx = 0; kx < 3; ++kx)
                acc += wk[ky * 3 + kx] * xp[(iy + ky) * W + ix + kx];
        s1[idx] = acc;
        ls += acc; lq += acc * acc;
    }
    reds[tid] = ls; redq[tid] = lq;
    __syncthreads();
    for (int s = 128; s > 0; s >>= 1) {
        if (tid < s) { reds[tid] += reds[tid + s]; redq[tid] += redq[tid + s]; }
        __syncthreads();
    }
    if (tid == 0) {
        const float cnt = 2.f * E1H * E1W;
        float m = reds[0] / cnt;
        float v = redq[0] / cnt - m * m;
        s_mean = m; s_rstd = rsqrtf(v + EPSF);
    }
    __syncthreads();
    for (int idx = tid; idx < 2 * E1H * E1W; idx += 256) {
        int oc = idx / (E1H * E1W);
        float vv = (s1[idx] - s_mean) * s_rstd * g_e1[2 * p + oc] + be_e1[2 * p + oc];
        s1[idx] = vv > 0.f ? vv : 0.f;
    }
    __syncthreads();

    // ---- stage e2: 2->4 ch, 3x3, stride (8,8) ----
    ls = 0.f; lq = 0.f;
    for (int idx = tid; idx < 4 * 36; idx += 256) {
        int oc = idx / 36;
        int r  = idx % 36;
        int oy = r / 6, ox = r % 6;
        const float* wk = w_e2 + ((4 * p + oc) * 2) * 9;
        float acc = b_e2[4 * p + oc];
        #pragma unroll
        for (int ic = 0; ic < 2; ++ic)
            #pragma unroll
            for (int ky = 0; ky < 3; ++ky)
                #pragma unroll
                for (int kx = 0; kx < 3; ++kx)
                    acc += wk[ic * 9 + ky * 3 + kx] *
                           s1[ic * (E1H * E1W) + (oy * 8 + ky) * E1W + ox * 8 + kx];
        s2[idx] = acc;
        ls += acc; lq += acc * acc;
    }
    reds[tid] = ls; redq[tid] = lq;
    __syncthreads();
    for (int s = 128; s > 0; s >>= 1) {
        if (tid < s) { reds[tid] += reds[tid + s]; redq[tid] += redq[tid + s]; }
        __syncthreads();
    }
    if (tid == 0) {
        float m = reds[0] / 144.f;
        float v = redq[0] / 144.f - m * m;
        s_mean = m; s_rstd = rsqrtf(v + EPSF);
    }
    __syncthreads();
    for (int idx = tid; idx < 144; idx += 256) {
        int oc = idx / 36;
        float vv = (s2[idx] - s_mean) * s_rstd * g_e2[4 * p + oc] + be_e2[4 * p + oc];
        s2[idx] = vv > 0.f ? vv : 0.f;
    }
    __syncthreads();

    // ---- stage e3: 4->8 ch, 3x3, stride (5,6) -> 1x1 ----
    if (tid < 8) {
        const float* wk = w_e3 + ((8 * p + tid) * 4) * 9;
        float acc = b_e3[8 * p + tid];
        #pragma unroll
        for (int ic = 0; ic < 4; ++ic)
            #pragma unroll
            for (int ky = 0; ky < 3; ++ky)
                #pragma unroll
                for (int kx = 0; kx < 3; ++kx)
                    acc += wk[ic * 9 + ky * 3 + kx] * s2[ic * 36 + ky * 6 + kx];
        s3[tid] = acc;
    }
    __syncthreads();
    if (tid == 0) {
        float m = 0.f, q = 0.f;
        for (int j = 0; j < 8; ++j) { m += s3[j]; q += s3[j] * s3[j]; }
        m *= 0.125f;
        float v = q * 0.125f - m * m;
        float rs = rsqrtf(v + EPSF);
        for (int j = 0; j < 8; ++j) {
            float vv = (s3[j] - m) * rs * g_e3[8 * p + j] + be_e3[8 * p + j];
            s3[j] = vv > 0.f ? vv : 0.f;
        }
    }
    __syncthreads();

    // ---- linear head: wb[j] = b_lin[j] + feat . W_lin[j,:] ----
    for (int j = tid; j < WB; j += 256) {
        float acc = b_lin[j];
        #pragma unroll
        for (int k = 0; k < 8; ++k) acc += s3[k] * W_lin[j * 8 + k];
        wb_ws[bp * WB + j] = acc;
    }
}

// ============================================================================
// Kernel 2: fused dynamic grouped conv: f = conv2(conv1(x)), both 3x3 with
// reflect-pad-1, 16 intermediate channels, per-panel dynamic weights.
// conv1 as WMMA GEMM: D[16ch x 16pix] = A[16 x 12taps] * B[12taps x 16pix]
// via 3x V_WMMA_F32_16X16X4_F32 (K padded 9->12). Interior tiles take a
// reflect-free fast path. All lane-half selects are hoisted as lane-constant
// integers so the chunk loops are branch-free (EXEC stays full for WMMA).
// ============================================================================
__global__ __launch_bounds__(128) void dynconv_kernel(
    const float* __restrict__ x, const float* __restrict__ wb,
    float* __restrict__ f)
{
    __shared__ float xs[XT * XT];        // 400  : reflect-mapped x tile
    __shared__ float gs[16 * NPIX];      // 5184 : conv1 output (16ch x 18x18)
    __shared__ float w1s[144], b1s[16], w2s[145];   // w2s[144] = b2

    const int x0  = blockIdx.x * TS;
    const int y0  = blockIdx.y * TS;
    const int pan = blockIdx.z;
    const int tid = threadIdx.x;
    const float* xp  = x + (size_t)pan * HW;
    const float* wbp = wb + pan * WB;

    for (int i = tid; i < 144; i += 128) w1s[i] = wbp[i];
    if (tid < 16) b1s[tid] = wbp[144 + tid];
    for (int i = tid; i < 145; i += 128) w2s[i] = wbp[160 + i];

    // stage x tile (halo 2, reflect-mapped so LDS content == padded image)
    for (int idx = tid; idx < XT * XT; idx += 128) {
        int i = idx / XT, j = idx % XT;
        stage_b32(xp + (size_t)refl(y0 - 2 + i, H) * W + refl(x0 - 2 + j, W),
                  &xs[idx]);
    }
    if (tid == 0) {
        // hint the next tile-row into the caches (global_prefetch_b8)
        __builtin_prefetch(xp + (size_t)refl(y0 + TS, H) * W + x0, 0, 1);
    }
    stage_wait();
    __syncthreads();

    const int lane = tid & 31;
    const int wv   = tid >> 5;           // wave id 0..3
    const int hi   = lane >> 4;          // lane half (selects K pair / M half)
    const int col  = lane & 15;

#if USE_WMMA4
    // -------- lane-constant operands hoisted out of the chunk loop --------
    // taps handled by this lane half: t00=2hi, t01=2hi+1, t10=4+2hi,
    // t11=5+2hi, t2=8 (lower half only); (dy,dx) = (t/3, t%3):
    const int dy00 = 0,      dx00 = 2 * hi;
    const int dy01 = hi,     dx01 = 1 - hi;
    const int dy10 = 1 + hi, dx10 = 1 - hi;
    const int dy11 = 1 + hi, dx11 = 2 - hi;

    v2f a0, a1, a2;
    a0.x = w1s[col * 9 + 2 * hi];         a0.y = w1s[col * 9 + 2 * hi + 1];
    a1.x = w1s[col * 9 + 4 + 2 * hi];     a1.y = w1s[col * 9 + 4 + 2 * hi + 1];
    {
        float w8 = w1s[col * 9 + 8];      // unconditional load, then select
        a2.x = hi ? 0.f : w8;
        a2.y = 0.f;
    }
    v8f bias;
    #pragma unroll
    for (int r = 0; r < 8; ++r) bias[r] = b1s[r + 8 * hi];

    const bool interior = (y0 >= 2) && (y0 + TS + 2 <= H) &&
                          (x0 >= 2) && (x0 + TS + 2 <= W);
    if (interior) {
        const int o00 = dy00 * XT + dx00;
        const int o01 = dy01 * XT + dx01;
        const int o10 = dy10 * XT + dx10;
        const int o11 = dy11 * XT + dx11;
        for (int c0 = wv; c0 < NCHK; c0 += 4) {
            int pix  = c0 * 16 + col;
            int pixc = pix < NPIX ? pix : NPIX - 1;
            int gy = pixc / GT, gx = pixc % GT;
            int base = gy * XT + gx;
            v2f b;
            v8f acc = bias;
            b.x = xs[base + o00];  b.y = xs[base + o01];
            acc = __builtin_amdgcn_wmma_f32_16x16x4_f32(false, a0, false, b,
                                                        (short)0, acc, false, false);
            b.x = xs[base + o10];  b.y = xs[base + o11];
            acc = __builtin_amdgcn_wmma_f32_16x16x4_f32(false, a1, false, b,
                                                        (short)0, acc, false, false);
            float t8 = xs[base + 2 * XT + 2];   // unconditional, then select
            b.x = hi ? 0.f : t8;
            b.y = 0.f;
            acc = __builtin_amdgcn_wmma_f32_16x16x4_f32(false, a2, false, b,
                                                        (short)0, acc, false, false);
            if (pix < NPIX) {
                #pragma unroll
                for (int r = 0; r < 8; ++r)
                    gs[(r + 8 * hi) * NPIX + pix] = acc[r];
            }
        }
    } else {
        for (int c0 = wv; c0 < NCHK; c0 += 4) {
            int pix  = c0 * 16 + col;
            int pixc = pix < NPIX ? pix : NPIX - 1;
            int gy = pixc / GT, gx = pixc % GT;
            int gyr = refl(y0 - 1 + gy, H);            // reflect g coordinate
            int gxr = refl(x0 - 1 + gx, W);
            // straight-line taps with lane-constant (dy,dx) — no divergence
            auto tap = [&](int dy, int dx) -> float {  // conv1 tap, x reflected
                int li = clampi(refl(gyr - 1 + dy, H) - (y0 - 2), 0, XT - 1);
                int lj = clampi(refl(gxr - 1 + dx, W) - (x0 - 2), 0, XT - 1);
                return xs[li * XT + lj];
            };
            v2f b;
            v8f acc = bias;
            b.x = tap(dy00, dx00);
            b.y = tap(dy01, dx01);
            acc = __builtin_amdgcn_wmma_f32_16x16x4_f32(false, a0, false, b,
                                                        (short)0, acc, false, false);
            b.x = tap(dy10, dx10);
            b.y = tap(dy11, dx11);
            acc = __builtin_amdgcn_wmma_f32_16x16x4_f32(false, a1, false, b,
                                                        (short)0, acc, false, false);
            float t8 = tap(2, 2);               // unconditional, then select
            b.x = hi ? 0.f : t8;
            b.y = 0.f;
            acc = __builtin_amdgcn_wmma_f32_16x16x4_f32(false, a2, false, b,
                                                        (short)0, acc, false, false);
            if (pix < NPIX) {
                #pragma unroll
                for (int r = 0; r < 8; ++r)
                    gs[(r + 8 * hi) * NPIX + pix] = acc[r];
            }
        }
    }
#else
    // VALU fallback (same math, plain FMAs)
    for (int idx = tid; idx < 16 * NPIX; idx += 128) {
        int ch = idx / NPIX, pix = idx % NPIX;
        int gy = pix / GT, gx = pix % GT;
        int gyr = refl(y0 - 1 + gy, H), gxr = refl(x0 - 1 + gx, W);
        float acc = b1s[ch];
        #pragma unroll
        for (int t = 0; t < 9; ++t) {
            int dy = t / 3, dx = t % 3;
            int li = clampi(refl(gyr - 1 + dy, H) - (y0 - 2), 0, XT - 1);
            int lj = clampi(refl(gxr - 1 + dx, W) - (x0 - 2), 0, XT - 1);
            acc += w1s[ch * 9 + t] * xs[li * XT + lj];
        }
        gs[idx] = acc;
    }
#endif
    __syncthreads();

    // conv2: 16ch x 3x3 -> 1; two adjacent pixels per thread (g-row reuse)
    {
        int py = tid >> 3;               // 0..15
        int px = (tid & 7) * 2;          // 0,2,..,14
        float acc0 = w2s[144];           // b2
        float acc1 = acc0;
        #pragma unroll 4
        for (int ch = 0; ch < 16; ++ch) {
            const float* g  = gs  + ch * NPIX;
            const float* wk = w2s + ch * 9;
            #pragma unroll
            for (int dy = 0; dy < 3; ++dy) {
                const float* gr = g + (py + dy) * GT + px;
                float g0 = gr[0], g1 = gr[1], g2 = gr[2], g3 = gr[3];
                float wa = wk[dy * 3], wbv = wk[dy * 3 + 1], wc = wk[dy * 3 + 2];
                acc0 += wa * g0 + wbv * g1 + wc * g2;
                acc1 += wa * g1 + wbv * g2 + wc * g3;
            }
        }
        int yy = y0 + py, xx = x0 + px;
        if (yy < H) {
            float* dst = f + (size_t)pan * HW + (size_t)yy * W + xx;
            if (xx < W)     dst[0] = acc0;
            if (xx + 1 < W) dst[1] = acc1;
        }
    }
}

// ============================================================================
// Kernel 3: shared peak-finding CNN (pf1: 1->6ch + BNeval + ReLU,
// pf2: 6->1ch + BNeval + ReLU) + residual + per-panel affine. Tiled, f staged
// through LDS with reflect mapping; pf2's reflect pad via double reflection.
// ============================================================================
__global__ __launch_bounds__(256) void peak_kernel(
    const float* __restrict__ f,
    const float* __restrict__ w_pf1, const float* __restrict__ b_pf1,
    const float* __restrict__ bn1_g, const float* __restrict__ bn1_b,
    const float* __restrict__ w_pf2, const float* __restrict__ b_pf2,
    const float* __restrict__ bn2_g, const float* __restrict__ bn2_b,
    const float* __restrict__ w_sc,  const float* __restrict__ b_sc,
    float* __restrict__ out)
{
    __shared__ float fs[XT * XT];         // 400
    __shared__ float y1[6 * NPIX];        // 1944
    __shared__ float wc1[54], wb1[6], sc1[6], sh1[6];
    __shared__ float wc2[54];
    __shared__ float sc2s, sh2s, b2s;

    const int x0  = blockIdx.x * TS;
    const int y0  = blockIdx.y * TS;
    const int pan = blockIdx.z;
    const int tid = threadIdx.x;
    const float* fp = f + (size_t)pan * HW;
    const float bn_rs = rsqrtf(1.f + EPSF);   // running_var = 1

    if (tid < 54) { wc1[tid] = w_pf1[tid]; wc2[tid] = w_pf2[tid]; }
    if (tid < 6)  { wb1[tid] = b_pf1[tid];
                    sc1[tid] = bn1_g[tid] * bn_rs;
                    sh1[tid] = bn1_b[tid]; }
    if (tid == 0) { sc2s = bn2_g[0] * bn_rs; sh2s = bn2_b[0]; b2s = b_pf2[0]; }

    for (int idx = tid; idx < XT * XT; idx += 256) {
        int i = idx / XT, j = idx % XT;
        stage_b32(fp + (size_t)refl(y0 - 2 + i, H) * W + refl(x0 - 2 + j, W),
                  &fs[idx]);
    }
    stage_wait();
    __syncthreads();

    // y1 = relu(bn1(conv_pf1(reflect_pad(f)))) on the 18x18 halo tile
    const bool interior = (y0 >= 2) && (y0 + TS + 2 <= H) &&
                          (x0 >= 2) && (x0 + TS + 2 <= W);
    if (interior) {
        for (int idx = tid; idx < 6 * NPIX; idx += 256) {
            int ch = idx / NPIX, pix = idx % NPIX;
            int gy = pix / GT, gx = pix % GT;
            int base = gy * XT + gx;
            float acc = wb1[ch];
            #pragma unroll
            for (int dy = 0; dy < 3; ++dy)
                #pragma unroll
                for (int dx = 0; dx < 3; ++dx)
                    acc += wc1[ch * 9 + dy * 3 + dx] * fs[base + dy * XT + dx];
            acc = acc * sc1[ch] + sh1[ch];
            y1[idx] = acc > 0.f ? acc : 0.f;
        }
    } else {
        for (int idx = tid; idx < 6 * NPIX; idx += 256) {
            int ch = idx / NPIX, pix = idx % NPIX;
            int gy = pix / GT, gx = pix % GT;
            int gyr = refl(y0 - 1 + gy, H), gxr = refl(x0 - 1 + gx, W);
            float acc = wb1[ch];
            #pragma unroll
            for (int t = 0; t < 9; ++t) {
                int dy = t / 3, dx = t % 3;
                int li = clampi(refl(gyr - 1 + dy, H) - (y0 - 2), 0, XT - 1);
                int lj = clampi(refl(gxr - 1 + dx, W) - (x0 - 2), 0, XT - 1);
                acc += wc1[ch * 9 + t] * fs[li * XT + lj];
            }
            acc = acc * sc1[ch] + sh1[ch];
            y1[idx] = acc > 0.f ? acc : 0.f;
        }
    }
    __syncthreads();

    // pf2 + BN + ReLU + residual + per-panel affine, one pixel per thread
    {
        int fy = tid >> 4, fx = tid & 15;
        float acc = b2s;
        #pragma unroll
        for (int ch = 0; ch < 6; ++ch)
            #pragma unroll
            for (int dy = 0; dy < 3; ++dy)
                #pragma unroll
                for (int dx = 0; dx < 3; ++dx)
                    acc += wc2[ch * 9 + dy * 3 + dx] *
                           y1[ch * NPIX + (fy + dy) * GT + fx + dx];
        acc = acc * sc2s + sh2s;
        acc = acc > 0.f ? acc : 0.f;
        acc += fs[(fy + 2) * XT + fx + 2];      // residual f
        int sci = pan & (NP - 1);               // panel index within sample
        acc = acc * w_sc[sci] + b_sc[sci];
        int yy = y0 + fy, xx = x0 + fx;
        if (yy < H && xx < W) out[(size_t)pan * HW + yy * W + xx] = acc;
    }
}

// ============================================================================
extern "C" void kernel_launch(void* const* d_in, const int* in_sizes, int n_in,
                              void* d_out, int out_size, void* d_ws, size_t ws_size,
                              hipStream_t stream)
{
    (void)in_sizes; (void)n_in; (void)out_size; (void)ws_size;
    const float* x     = (const float*)d_in[0];
    const float* w_e1  = (const float*)d_in[1];
    const float* b_e1  = (const float*)d_in[2];
    const float* g_e1  = (const float*)d_in[3];
    const float* be_e1 = (const float*)d_in[4];
    const float* w_e2  = (const float*)d_in[5];
    const float* b_e2  = (const float*)d_in[6];
    const float* g_e2  = (const float*)d_in[7];
    const float* be_e2 = (const float*)d_in[8];
    const float* w_e3  = (const float*)d_in[9];
    const float* b_e3  = (const float*)d_in[10];
    const float* g_e3  = (const float*)d_in[11];
    const float* be_e3 = (const float*)d_in[12];
    const float* W_lin = (const float*)d_in[13];
    const float* b_lin = (const float*)d_in[14];
    const float* w_pf1 = (const float*)d_in[15];
    const float* b_pf1 = (const float*)d_in[16];
    const float* bn1_g = (const float*)d_in[17];
    const float* bn1_b = (const float*)d_in[18];
    const float* w_pf2 = (const float*)d_in[19];
    const float* b_pf2 = (const float*)d_in[20];
    const float* bn2_g = (const float*)d_in[21];
    const float* bn2_b = (const float*)d_in[22];
    const float* w_sc  = (const float*)d_in[23];
    const float* b_sc  = (const float*)d_in[24];

    // workspace layout: wb[64*305] then f[64*185*388] (~18.5 MB total)
    float* wb_ws = (float*)d_ws;
    float* f_ws  = (float*)d_ws + 20480;   // 80 KB aligned start for f

    encoder_kernel<<<dim3(NPAN), dim3(256), 0, stream>>>(
        x, w_e1, b_e1, g_e1, be_e1, w_e2, b_e2, g_e2, be_e2,
        w_e3, b_e3, g_e3, be_e3, W_lin, b_lin, wb_ws);

    dynconv_kernel<<<dim3(TILES_X, TILES_Y, NPAN), dim3(128), 0, stream>>>(
        x, wb_ws, f_ws);

    peak_kernel<<<dim3(TILES_X, TILES_Y, NPAN), dim3(256), 0, stream>>>(
        f_ws, w_pf1, b_pf1, bn1_g, bn1_b, w_pf2, b_pf2, bn2_g, bn2_b,
        w_sc, b_sc, (float*)d_out);
}